// SimpleHGNConv_15135464751355
// MI455X (gfx1250) — compile-verified
//
#include <hip/hip_runtime.h>
#include <hip/hip_bf16.h>

#define IN_DIM     256
#define OUT_DIM    32
#define HEADS      4
#define OH         128           // OUT_DIM * HEADS
#define EDGE_DIM   64
#define NUM_ETYPES 8
#define NEG_SLOPE  0.2f

typedef float v2f __attribute__((ext_vector_type(2)));
typedef float v4f __attribute__((ext_vector_type(4)));
typedef float v8f __attribute__((ext_vector_type(8)));

// ---------------------------------------------------------------------------
// Kernel 1: fused GEMM  P = h @ [W | res_W]  using V_WMMA_F32_16X16X4_F32.
// One wave computes a 16-row x 128-col stripe (8 accumulators of 16x16).
// half==0 -> emb (with NaN->0), half==1 -> res (+ res_b).
// A-frag (16x4 f32, ISA 7.12.2): lane l holds m=l&15, k=(l>>4)*2 + j (j=0,1)
// B-frag (4x16 f32):             lane l holds n=l&15, k=(l>>4)*2 + j
// C/D   (16x16 f32, 8 VGPRs):    vgpr v, lane l -> m=v+(l>>4)*8, n=l&15
// ---------------------------------------------------------------------------
__global__ __launch_bounds__(128) void hgn_gemm_wmma(
    const float* __restrict__ h, const float* __restrict__ W,
    const float* __restrict__ resW, const float* __restrict__ res_b,
    float* __restrict__ emb, float* __restrict__ res, int n)
{
    const int lane    = threadIdx.x & 31;
    const int gwave   = (blockIdx.x * blockDim.x + threadIdx.x) >> 5;
    const int rowTile = gwave >> 1;
    const int half    = gwave & 1;
    const int row0    = rowTile * 16;
    if (row0 >= n) return;

    const float* __restrict__ B = half ? resW : W;

    const int m  = lane & 15;
    const int kk = (lane >> 4) << 1;          // 0 or 2

    int arow = row0 + m; if (arow >= n) arow = n - 1;
    const float* aptr = h + (size_t)arow * IN_DIM + kk;

    v8f acc[8];
#pragma unroll
    for (int t = 0; t < 8; ++t)
        acc[t] = (v8f){0.f,0.f,0.f,0.f,0.f,0.f,0.f,0.f};

    for (int k0 = 0; k0 < IN_DIM; k0 += 4) {
        v2f a = *(const v2f*)(aptr + k0);
        const float* bptr = B + (size_t)(k0 + kk) * OH + m;
#pragma unroll
        for (int t = 0; t < 8; ++t) {
            v2f b;
            b.x = bptr[t * 16];        // k = kk,   n = t*16 + m
            b.y = bptr[t * 16 + OH];   // k = kk+1, n = t*16 + m
            acc[t] = __builtin_amdgcn_wmma_f32_16x16x4_f32(
                false, a, false, b, (short)0, acc[t], false, false);
        }
    }

    const int rbase = row0 + ((lane >> 4) << 3);
#pragma unroll
    for (int t = 0; t < 8; ++t) {
        const int c = t * 16 + m;
#pragma unroll
        for (int v = 0; v < 8; ++v) {
            const int r = rbase + v;
            if (r >= n) continue;
            float val = acc[t][v];
            if (half == 0) {
                if (val != val) val = 0.f;               // jnp.where(isnan, 0, .)
                emb[(size_t)r * OH + c] = val;
            } else {
                res[(size_t)r * OH + c] = val + res_b[c];
            }
        }
    }
}

// ---------------------------------------------------------------------------
// Kernel 2: h_e_type[t][hd] = sum_f a_e[hd][f] * sum_e edge_emb[t][e]*Wr[t][e][hd*64+f]
// 8*4 = 32 outputs, one wave.
// ---------------------------------------------------------------------------
__global__ void hgn_etype(const float* __restrict__ edge_emb,
                          const float* __restrict__ Wr,
                          const float* __restrict__ a_e,
                          float* __restrict__ h_et)
{
    const int idx = threadIdx.x;
    if (idx >= NUM_ETYPES * HEADS) return;
    const int t = idx >> 2, hd = idx & 3;
    const float* ee = edge_emb + t * EDGE_DIM;
    float acc = 0.f;
    for (int f = 0; f < EDGE_DIM; ++f) {
        const float* wr = Wr + ((size_t)t * EDGE_DIM) * (EDGE_DIM * HEADS)
                             + hd * EDGE_DIM + f;
        float s = 0.f;
        for (int e = 0; e < EDGE_DIM; ++e)
            s += ee[e] * wr[(size_t)e * (EDGE_DIM * HEADS)];
        acc += a_e[hd * EDGE_DIM + f] * s;
    }
    h_et[idx] = acc;
}

// ---------------------------------------------------------------------------
// Kernel 3: h_l / h_r attention dots, one thread per (node, head).
// ---------------------------------------------------------------------------
__global__ void hgn_hlr(const float* __restrict__ emb,
                        const float* __restrict__ a_l,
                        const float* __restrict__ a_r,
                        float* __restrict__ h_l, float* __restrict__ h_r, int n)
{
    const int idx = blockIdx.x * blockDim.x + threadIdx.x;
    if (idx >= n * HEADS) return;
    const int node = idx >> 2, hd = idx & 3;
    const float* e  = emb + (size_t)node * OH + hd * OUT_DIM;
    const float* al = a_l + hd * OUT_DIM;
    const float* ar = a_r + hd * OUT_DIM;
    float sl = 0.f, sr = 0.f;
#pragma unroll 8
    for (int d = 0; d < OUT_DIM; ++d) { float v = e[d]; sl += al[d]*v; sr += ar[d]*v; }
    h_l[idx] = sl; h_r[idx] = sr;
}

// ---------------------------------------------------------------------------
// Kernel 4: init out (atomic accumulator), seg_max=-inf, seg_sum=0.
// ---------------------------------------------------------------------------
__global__ void hgn_init(float* __restrict__ out, float* __restrict__ seg_max,
                         float* __restrict__ seg_sum, int n)
{
    const int idx = blockIdx.x * blockDim.x + threadIdx.x;
    if (idx < n * OH) out[idx] = 0.f;
    if (idx < n * HEADS) { seg_max[idx] = -__builtin_inff(); seg_sum[idx] = 0.f; }
}

// Signed float atomic max via int-max / uint-min (init must be -inf).
__device__ __forceinline__ void atomicMaxF(float* addr, float v) {
    if (v >= 0.f) atomicMax((int*)addr, __float_as_int(v));
    else          atomicMin((unsigned int*)addr, __float_as_uint(v));
}

// ---------------------------------------------------------------------------
// Kernel 5: att = leaky_relu(h_l[row]+h_r[col]+h_et[etype]); seg_max by col.
// ---------------------------------------------------------------------------
__global__ void hgn_att_max(const float* __restrict__ h_l, const float* __restrict__ h_r,
                            const float* __restrict__ h_et, const int* __restrict__ row,
                            const int* __restrict__ col, const int* __restrict__ etype,
                            float* __restrict__ att, float* __restrict__ seg_max, int eCnt)
{
    const int e = blockIdx.x * blockDim.x + threadIdx.x;
    if (e >= eCnt) return;
    const int r = row[e], c = col[e], t = etype[e];
    const v4f hl = *(const v4f*)(h_l + (size_t)r * 4);
    const v4f hr = *(const v4f*)(h_r + (size_t)c * 4);
    const v4f he = *(const v4f*)(h_et + t * 4);
    v4f a;
#pragma unroll
    for (int hd = 0; hd < 4; ++hd) {
        float x = hl[hd] + hr[hd] + he[hd];
        x = x > 0.f ? x : NEG_SLOPE * x;
        a[hd] = x;
        atomicMaxF(seg_max + (size_t)c * 4 + hd, x);
    }
    *(v4f*)(att + (size_t)e * 4) = a;
}

// ---------------------------------------------------------------------------
// Kernel 6: att <- exp(att - seg_max[col]); seg_sum by col.
// ---------------------------------------------------------------------------
__global__ void hgn_exp_sum(const int* __restrict__ col, float* __restrict__ att,
                            const float* __restrict__ seg_max,
                            float* __restrict__ seg_sum, int eCnt)
{
    const int e = blockIdx.x * blockDim.x + threadIdx.x;
    if (e >= eCnt) return;
    const int c = col[e];
    v4f a        = *(v4f*)(att + (size_t)e * 4);
    const v4f mx = *(const v4f*)(seg_max + (size_t)c * 4);
#pragma unroll
    for (int hd = 0; hd < 4; ++hd) {
        float ex = expf(a[hd] - mx[hd]);
        a[hd] = ex;
        atomicAdd(seg_sum + (size_t)c * 4 + hd, ex);
    }
    *(v4f*)(att + (size_t)e * 4) = a;
}

// ---------------------------------------------------------------------------
// Kernel 7: aggregation, one wave per edge. Lane l gathers emb[row] channels
// 4l..4l+3 (one coalesced float4) and does 4 L2 atomics into out[col] at the
// permuted position d*4+h  (emb channel ch = h*32+d,  out channel = d*4+h).
// ---------------------------------------------------------------------------
__global__ __launch_bounds__(128) void hgn_agg(
    const float* __restrict__ emb, const float* __restrict__ att,
    const float* __restrict__ seg_sum, const int* __restrict__ row,
    const int* __restrict__ col, float* __restrict__ out, int eCnt)
{
    const int e = (blockIdx.x * blockDim.x + threadIdx.x) >> 5;
    if (e >= eCnt) return;
    const int lane = threadIdx.x & 31;
    const int r = row[e], c = col[e];
    const v4f ex = *(const v4f*)(att + (size_t)e * 4);
    const v4f ss = *(const v4f*)(seg_sum + (size_t)c * 4);
    v4f alpha;
#pragma unroll
    for (int hd = 0; hd < 4; ++hd) alpha[hd] = ex[hd] / ss[hd];

    const v4f ev = *(const v4f*)(emb + (size_t)r * OH + lane * 4);
    float* ob = out + (size_t)c * OH;
#pragma unroll
    for (int j = 0; j < 4; ++j) {
        const int ch = lane * 4 + j;          // = hd*32 + d
        const int hd = ch >> 5, d = ch & 31;
        atomicAdd(ob + d * 4 + hd, ev[j] * alpha[hd]);
    }
}

// ---------------------------------------------------------------------------
// Kernel 8: out = elu(agg + res), vectorized float4.
// ---------------------------------------------------------------------------
__global__ void hgn_final(const float* __restrict__ res, float* __restrict__ out, int n)
{
    const int idx = blockIdx.x * blockDim.x + threadIdx.x;
    if (idx >= n * (OH / 4)) return;
    v4f a       = *(v4f*)(out + (size_t)idx * 4);
    const v4f b = *(const v4f*)(res + (size_t)idx * 4);
#pragma unroll
    for (int j = 0; j < 4; ++j) {
        float x = a[j] + b[j];
        a[j] = x > 0.f ? x : expm1f(x);
    }
    *(v4f*)(out + (size_t)idx * 4) = a;
}

// ---------------------------------------------------------------------------
extern "C" void kernel_launch(void* const* d_in, const int* in_sizes, int n_in,
                              void* d_out, int out_size, void* d_ws, size_t ws_size,
                              hipStream_t stream)
{
    const float* h    = (const float*)d_in[0];
    const float* W    = (const float*)d_in[1];
    const float* eemb = (const float*)d_in[2];
    const float* Wr   = (const float*)d_in[3];
    const float* a_l  = (const float*)d_in[4];
    const float* a_r  = (const float*)d_in[5];
    const float* a_e  = (const float*)d_in[6];
    const float* resW = (const float*)d_in[7];
    const float* resb = (const float*)d_in[8];
    const int*   row  = (const int*)d_in[9];
    const int*   col  = (const int*)d_in[10];
    const int*   et   = (const int*)d_in[11];
    float* out = (float*)d_out;

    const int n = in_sizes[0] / IN_DIM;
    const int E = in_sizes[9];

    char* ws = (char*)d_ws;
    size_t off = 0;
    auto carve = [&](size_t bytes) -> float* {
        float* p = (float*)(ws + off);
        off += (bytes + 255) & ~(size_t)255;
        return p;
    };
    float* emb   = carve((size_t)n * OH * sizeof(float));
    float* res   = carve((size_t)n * OH * sizeof(float));
    float* h_l   = carve((size_t)n * HEADS * sizeof(float));
    float* h_r   = carve((size_t)n * HEADS * sizeof(float));
    float* h_et  = carve((size_t)NUM_ETYPES * HEADS * sizeof(float));
    float* segmx = carve((size_t)n * HEADS * sizeof(float));
    float* segsm = carve((size_t)n * HEADS * sizeof(float));
    float* att   = carve((size_t)E * HEADS * sizeof(float));
    (void)ws_size; (void)n_in; (void)out_size;

    const int rowTiles = (n + 15) / 16;
    const int waves    = rowTiles * 2;          // two 128-col halves per row tile

    hgn_gemm_wmma<<<(waves + 3) / 4, 128, 0, stream>>>(h, W, resW, resb, emb, res, n);
    hgn_etype   <<<1, 32, 0, stream>>>(eemb, Wr, a_e, h_et);
    hgn_hlr     <<<(n * HEADS + 255) / 256, 256, 0, stream>>>(emb, a_l, a_r, h_l, h_r, n);
    hgn_init    <<<(n * OH + 255) / 256, 256, 0, stream>>>(out, segmx, segsm, n);
    hgn_att_max <<<(E + 255) / 256, 256, 0, stream>>>(h_l, h_r, h_et, row, col, et, att, segmx, E);
    hgn_exp_sum <<<(E + 255) / 256, 256, 0, stream>>>(col, att, segmx, segsm, E);
    hgn_agg     <<<(E + 3) / 4, 128, 0, stream>>>(emb, att, segsm, row, col, out, E);
    hgn_final   <<<(n * (OH / 4) + 255) / 256, 256, 0, stream>>>(res, out, n);
}